// GCN_53257594470567
// MI455X (gfx1250) — compile-verified
//
#include <hip/hip_runtime.h>
#include <math.h>

// Problem constants (fixed by the reference).
#define NN 100000      // nodes
#define INC 128        // input channels
#define HIDC 64        // hidden channels
#define OUTC 40        // output channels
#define OUTCP 48       // OUTC padded to a 16-column tile multiple
#define NE 1600000     // edges

typedef float v2f __attribute__((ext_vector_type(2)));
typedef float v8f __attribute__((ext_vector_type(8)));

// ---------------- degree / normalization ----------------

__global__ void k_init_deg(float* __restrict__ deg, int n) {
    int i = blockIdx.x * blockDim.x + threadIdx.x;
    if (i < n) deg[i] = 1.0f;  // self loop
}

__global__ void k_deg_scatter(const long long* __restrict__ dst,
                              float* __restrict__ deg, int e) {
    int i = blockIdx.x * blockDim.x + threadIdx.x;
    if (i < e) unsafeAtomicAdd(&deg[(int)dst[i]], 1.0f);
}

__global__ void k_rsqrt_inplace(float* __restrict__ d, int n) {
    int i = blockIdx.x * blockDim.x + threadIdx.x;
    if (i < n) d[i] = rsqrtf(d[i]);
}

// ---------------- pad W2 (HIDC x OUTC) -> (HIDC x OUTCP) with zeros ---------

__global__ void k_pad_w2(const float* __restrict__ W2, float* __restrict__ W2p) {
    int i = blockIdx.x * blockDim.x + threadIdx.x;
    if (i >= HIDC * OUTCP) return;
    int r = i / OUTCP;
    int c = i - r * OUTCP;
    W2p[i] = (c < OUTC) ? W2[r * OUTC + c] : 0.0f;
}

// ---------------- WMMA f32 GEMM: C[M x NC] = op(A)[M x K] * B[K x LDB] ------
// One wave computes one 16x16 output tile with V_WMMA_F32_16X16X4_F32.
// K, LDB, NC are compile-time so all fragment loads strength-reduce to a
// single base pointer + constant immediate offsets, and the K loop fully
// unrolls with no per-iteration address math or exec masking.
// Row tile from blockIdx.x*waves+wave (wave-uniform guard), column tile from
// blockIdx.y. LDB is a multiple of 16 (B pre-padded), so B loads are
// unconditional; the C store is guarded only when NC % 16 != 0.

template <int K, int LDB, int NC, bool RELU>
__global__ void k_gemm_wmma_f32(const float* __restrict__ A,
                                const float* __restrict__ B,
                                float* __restrict__ C) {
    const int lane = threadIdx.x & 31;
    const int wave = threadIdx.x >> 5;
    const int tm = blockIdx.x * (blockDim.x >> 5) + wave;
    if (tm >= NN / 16) return;  // wave-uniform

    const int half = lane >> 4;   // 0: K pair {k,k+1}, 1: {k+2,k+3}
    const int l    = lane & 15;
    const int col  = (blockIdx.y << 4) + l;   // always < LDB

    const float* __restrict__ arow = A + (size_t)(tm * 16 + l) * K + half * 2;
    const float* __restrict__ bcol = B + (size_t)(half * 2) * LDB + col;

    v8f acc = {};
#pragma unroll
    for (int k = 0; k < K; k += 4) {
        // A fragment (16x4 f32): 8-byte aligned pair -> global_load_b64.
        v2f av = *(const v2f*)(arow + k);
        if (RELU) { av.x = fmaxf(av.x, 0.0f); av.y = fmaxf(av.y, 0.0f); }

        // B fragment (4x16 f32): two b32 loads at constant offsets.
        v2f bv = { bcol[(size_t)k * LDB], bcol[(size_t)(k + 1) * LDB] };

        acc = __builtin_amdgcn_wmma_f32_16x16x4_f32(
            /*neg_a=*/false, av, /*neg_b=*/false, bv,
            /*c_mod=*/(short)0, acc, /*reuse_a=*/false, /*reuse_b=*/false);
    }

    if ((NC % 16 == 0) || (col < NC)) {
        const int rbase = tm * 16 + half * 8;
#pragma unroll
        for (int r = 0; r < 8; ++r)
            C[(size_t)(rbase + r) * NC + col] = acc[r];
    }
}

// ---------------- self-loop init: agg = h * dis^2 + bias ----------------

template <int C>
__global__ void k_self_init(const float* __restrict__ h,
                            const float* __restrict__ dis,
                            const float* __restrict__ bias,
                            float* __restrict__ agg, int n) {
    size_t t = (size_t)blockIdx.x * blockDim.x + threadIdx.x;
    if (t >= (size_t)n * C) return;
    int row = (int)(t / C);
    int c   = (int)(t - (size_t)row * C);
    float di = dis[row];
    agg[t] = h[t] * di * di + bias[c];
}

// ---------------- edge scatter: agg[dst] += dis[src]*dis[dst]*h[src] --------
// TPE = C/4 threads per edge, each handling a float4 of channels.

template <int C, int TPE>
__global__ void k_edge_scatter(const long long* __restrict__ src,
                               const long long* __restrict__ dst,
                               const float* __restrict__ dis,
                               const float* __restrict__ h,
                               float* __restrict__ agg, int e) {
    size_t t = (size_t)blockIdx.x * blockDim.x + threadIdx.x;
    if (t >= (size_t)e * TPE) return;
    int ed = (int)(t / TPE);
    int cg = (int)(t - (size_t)ed * TPE) * 4;
    int s = (int)src[ed];
    int d = (int)dst[ed];
    float w = dis[s] * dis[d];
    const float4 hv = *(const float4*)(h + (size_t)s * C + cg);
    float* __restrict__ o = agg + (size_t)d * C + cg;
    unsafeAtomicAdd(o + 0, w * hv.x);
    unsafeAtomicAdd(o + 1, w * hv.y);
    unsafeAtomicAdd(o + 2, w * hv.z);
    unsafeAtomicAdd(o + 3, w * hv.w);
}

// ---------------- row-wise log_softmax over OUTC=40, one wave per row -------

__global__ void k_log_softmax(float* __restrict__ out, int n) {
    const int lane = threadIdx.x & 31;
    const int wave = threadIdx.x >> 5;
    const int row  = blockIdx.x * (blockDim.x >> 5) + wave;
    if (row >= n) return;
    float* __restrict__ p = out + (size_t)row * OUTC;

    float v0 = p[lane];                                        // cols 0..31
    float v1 = (lane < OUTC - 32) ? p[32 + lane] : -INFINITY;  // cols 32..39

    float m = fmaxf(v0, v1);
#pragma unroll
    for (int o = 16; o > 0; o >>= 1) m = fmaxf(m, __shfl_xor(m, o, 32));

    float s = expf(v0 - m) + ((lane < OUTC - 32) ? expf(v1 - m) : 0.0f);
#pragma unroll
    for (int o = 16; o > 0; o >>= 1) s += __shfl_xor(s, o, 32);

    float lse = m + logf(s);
    p[lane] = v0 - lse;
    if (lane < OUTC - 32) p[32 + lane] = v1 - lse;
}

// ---------------- orchestration ----------------

extern "C" void kernel_launch(void* const* d_in, const int* in_sizes, int n_in,
                              void* d_out, int out_size, void* d_ws, size_t ws_size,
                              hipStream_t stream) {
    (void)in_sizes; (void)n_in; (void)out_size; (void)ws_size;

    const float*     x   = (const float*)d_in[0];
    const long long* ei  = (const long long*)d_in[1];  // (2, E) int64
    const float*     W1  = (const float*)d_in[2];
    const float*     b1  = (const float*)d_in[3];
    const float*     W2  = (const float*)d_in[4];
    const float*     b2  = (const float*)d_in[5];
    float*           out = (float*)d_out;

    const long long* e_src = ei;
    const long long* e_dst = ei + NE;

    // Workspace carve-up (all 16B-aligned sizes):
    //   dis  : N floats              (0.4 MB)
    //   h1   : N*64 floats           (25.6 MB)  -- later reused as h2 (N*40)
    //   agg1 : N*64 floats           (25.6 MB)
    //   w2p  : 64*48 floats          (12 KB)    -- zero-padded W2
    char* ws    = (char*)d_ws;
    float* dis  = (float*)ws;
    float* h1   = (float*)(ws + (size_t)NN * sizeof(float));
    float* agg1 = h1 + (size_t)NN * HIDC;
    float* w2p  = agg1 + (size_t)NN * HIDC;
    float* h2   = h1;  // h1 is dead once agg1 is built; alias to save memory

    const int B = 256;
    const int WPB = B / 32;                      // 8 waves per block
    const int rowTileBlocks = (NN / 16 + WPB - 1) / WPB;  // 782

    // 1) deg = 1 + in-degree; dis = rsqrt(deg)
    k_init_deg<<<(NN + B - 1) / B, B, 0, stream>>>(dis, NN);
    k_deg_scatter<<<(NE + B - 1) / B, B, 0, stream>>>(e_dst, dis, NE);
    k_rsqrt_inplace<<<(NN + B - 1) / B, B, 0, stream>>>(dis, NN);

    // Pad W2 into workspace (independent of the above; just queued in order).
    k_pad_w2<<<(HIDC * OUTCP + B - 1) / B, B, 0, stream>>>(W2, w2p);

    // 2) h1 = x @ W1   (100000x128 @ 128x64), WMMA f32
    k_gemm_wmma_f32<INC, HIDC, HIDC, false>
        <<<dim3(rowTileBlocks, HIDC / 16), B, 0, stream>>>(x, W1, h1);

    // 3) agg1 = dis^2 * h1 + b1 ; scatter edge messages ; (ReLU fused later)
    k_self_init<HIDC><<<(int)(((size_t)NN * HIDC + B - 1) / B), B, 0, stream>>>(
        h1, dis, b1, agg1, NN);
    k_edge_scatter<HIDC, HIDC / 4><<<(int)(((size_t)NE * (HIDC / 4) + B - 1) / B),
                                     B, 0, stream>>>(e_src, e_dst, dis, h1, agg1, NE);

    // 4) h2 = relu(agg1) @ W2pad  (100000x64 @ 64x48), WMMA f32, ReLU fused,
    //    stores guarded to the real 40 columns.
    k_gemm_wmma_f32<HIDC, OUTCP, OUTC, true>
        <<<dim3(rowTileBlocks, OUTCP / 16), B, 0, stream>>>(agg1, w2p, h2);

    // 5) layer-2 aggregation straight into d_out
    k_self_init<OUTC><<<(int)(((size_t)NN * OUTC + B - 1) / B), B, 0, stream>>>(
        h2, dis, b2, out, NN);
    k_edge_scatter<OUTC, OUTC / 4><<<(int)(((size_t)NE * (OUTC / 4) + B - 1) / B),
                                     B, 0, stream>>>(e_src, e_dst, dis, h2, out, NE);

    // 6) log_softmax rows in place
    k_log_softmax<<<(NN + (B / 32) - 1) / (B / 32), B, 0, stream>>>(out, NN);
}